// Encoder_12240656794040
// MI455X (gfx1250) — compile-verified
//
#include <hip/hip_runtime.h>

typedef __attribute__((ext_vector_type(2))) float v2f;
typedef __attribute__((ext_vector_type(8))) float v8f;

#define FEAT   128
#define TWOF   256
#define EMBED  128
#define NSAMP  10

// One block per batch element. 256 threads = 8 waves (wave32).
// Phase 1: threads 0..127 gather self feature row; threads 128..255 compute
//          neighbor mean -> combined[256] in LDS.
// Phase 2: wave w owns output columns [16w, 16w+16). 64 chunks of K=4 via
//          V_WMMA_F32_16X16X4_F32 with broadcast-A (all 16 rows identical),
//          so every D row holds the true 1x256 . 256x16 result.
__global__ __launch_bounds__(256)
void sage_encoder_kernel(const float* __restrict__ features,
                         const float* __restrict__ weight,
                         const int*   __restrict__ nodes,
                         const int*   __restrict__ neigh,
                         float*       __restrict__ out)
{
    __shared__ float comb[TWOF];

    const int b = blockIdx.x;
    const int t = threadIdx.x;

    if (t < FEAT) {
        // self embedding row, coalesced 512B across the half-block
        const long si = (long)nodes[b] * FEAT + t;
        comb[t] = features[si];
    } else {
        // mean over 10 sampled neighbor rows
        const int f = t - FEAT;
        float s = 0.0f;
        #pragma unroll
        for (int k = 0; k < NSAMP; ++k) {
            const long ni = (long)neigh[b * NSAMP + k] * FEAT + f;
            s += features[ni];
        }
        comb[FEAT + f] = s * (1.0f / (float)NSAMP);
    }
    __syncthreads();

    const int wave = t >> 5;           // 0..7 -> 16-column output tile
    const int lane = t & 31;
    const int n    = lane & 15;        // column within tile
    const int klo  = (lane >> 4) << 1; // 0 for lanes 0-15, 2 for lanes 16-31
    const int e0   = wave << 4;

    // Per-lane base pointer into this batch element's weight matrix:
    // W[b][row = klo][col = e0 + n]; chunk tk advances rows by 4.
    const float* __restrict__ wbase =
        weight + (long)b * (TWOF * EMBED) + (long)klo * EMBED + (e0 + n);
    const float* abase = comb + klo;

    v8f acc = {};
    #pragma unroll 8
    for (int tk = 0; tk < TWOF / 4; ++tk) {
        // A (broadcast rows): lane supplies comb[4*tk+klo], comb[4*tk+klo+1]
        v2f a = *(const v2f*)(abase + 4 * tk);          // ds_load_b64
        // B: rows 4*tk+klo and 4*tk+klo+1, column e0+n
        v2f bb;
        bb.x = wbase[(long)(4 * tk)     * EMBED];       // global_load_b32
        bb.y = wbase[(long)(4 * tk + 1) * EMBED];       // global_load_b32
        // D = A x B + C   (v_wmma_f32_16x16x4_f32)
        acc = __builtin_amdgcn_wmma_f32_16x16x4_f32(
            /*neg_a=*/false, a, /*neg_b=*/false, bb,
            /*c_mod=*/(short)0, acc, /*reuse_a=*/false, /*reuse_b=*/false);
    }

    // All D rows are identical (broadcast A). Row 0 for columns 0..15 lives in
    // acc[0] of lanes 0..15 (lanes 16..31 hold row 8 == row 0, redundant).
    if (lane < 16) {
        float v = acc[0];
        out[(long)b * EMBED + e0 + n] = v > 0.0f ? v : 0.0f;
    }
}

extern "C" void kernel_launch(void* const* d_in, const int* in_sizes, int n_in,
                              void* d_out, int out_size, void* d_ws, size_t ws_size,
                              hipStream_t stream) {
    (void)n_in; (void)d_ws; (void)ws_size; (void)out_size;
    const float* features = (const float*)d_in[0];
    const float* weight   = (const float*)d_in[1];
    const int*   nodes    = (const int*)d_in[2];
    const int*   neigh    = (const int*)d_in[3];
    float*       out      = (float*)d_out;

    const int batch = in_sizes[2];  // 4096
    sage_encoder_kernel<<<batch, 256, 0, stream>>>(features, weight, nodes, neigh, out);
}